// RayMarchingGenerateInsideLayer_90271622627608
// MI455X (gfx1250) — compile-verified
//
#include <hip/hip_runtime.h>

// RayMarching fused kernel for gfx1250 (MI455X).
//   q = r @ W1 computed ONCE via V_WMMA_F32_16X16X4_F32 into registers
//   (norm folded into the alpha recurrence), then n_iter register-only
//   elementwise iterations, then coalesced float4 output.

#define D_DIM 128
#define H_DIM 256
#define ROWS_PER_BLOCK 128
#define THREADS 256
#define W1T_STRIDE 130   // 128 + 2 pad floats -> conflict-free ds_load_b64

typedef float v2f __attribute__((ext_vector_type(2)));
typedef float v8f __attribute__((ext_vector_type(8)));

__device__ __forceinline__ float fast_tanh(float x) {
    // tanh(x) = 1 - 2/(exp(2x)+1); saturates correctly at +-inf
    float e = __expf(2.0f * x);
    return 1.0f - __fdividef(2.0f, e + 1.0f);
}

// p1[j] = b1[j] + sum_k pivot[k] * W1[k][j]   (tiny: 256 outputs x 128 MACs)
__global__ void rm_p1_kernel(const float* __restrict__ pivot,
                             const float* __restrict__ W1,
                             const float* __restrict__ b1,
                             float* __restrict__ p1) {
    int j = threadIdx.x;
    float acc = b1[j];
#pragma unroll 4
    for (int k = 0; k < D_DIM; ++k)
        acc = fmaf(pivot[k], W1[k * H_DIM + j], acc);
    p1[j] = acc;
}

__launch_bounds__(THREADS, 1)
__global__ void rm_main_kernel(const float* __restrict__ r,
                               const float* __restrict__ s,
                               const float* __restrict__ pivot,
                               const float* __restrict__ W1,
                               const float* __restrict__ w2,
                               const float* __restrict__ b2,
                               const int* __restrict__ n_iter_p,
                               const float* __restrict__ p1,
                               float* __restrict__ out) {
    __shared__ float lds_w1t[H_DIM * W1T_STRIDE];          // 133,120 B  W1^T padded
    __shared__ __align__(16) float lds_pivot[D_DIM];
    __shared__ float lds_f[ROWS_PER_BLOCK];

    const int tid    = threadIdx.x;
    const int wave   = tid >> 5;
    const int lane   = tid & 31;
    const int lane16 = lane & 15;
    const int hi     = lane >> 4;    // 0: lanes 0-15 (K%4 in {0,1}), 1: lanes 16-31 (K%4 in {2,3})

    // ---- stage W1 transposed into LDS: lds_w1t[n][k] = W1[k][n] (coalesced global reads,
    //      conflict-free LDS writes: stride 130 == 2 mod 64)
    for (int k = 0; k < D_DIM; ++k)
        lds_w1t[tid * W1T_STRIDE + k] = W1[k * H_DIM + tid];
    if (tid < D_DIM) lds_pivot[tid] = pivot[tid];

    // ---- row norms (float4 pass, independent of the GEMM fragment loads)
    const int rowBase = blockIdx.x * ROWS_PER_BLOCK + wave * 16;
    const float4* rrow4 = (const float4*)(r + (size_t)(rowBase + lane16) * D_DIM);
    float ss = 0.0f;
#pragma unroll
    for (int i = 0; i < 16; ++i) {          // lo lanes: even float4s, hi lanes: odd float4s
        float4 v = rrow4[2 * i + hi];
        ss = fmaf(v.x, v.x, fmaf(v.y, v.y, fmaf(v.z, v.z, fmaf(v.w, v.w, ss))));
    }
    ss += __shfl_xor(ss, 16);               // full row sum-of-squares, replicated in both halves
    const float rinv = rsqrtf(ss);
    // move rinv into the C-tile row layout: rinv_a[j] = 1/||r_row||, row = j + 8*hi
    float rinv_a[8];
#pragma unroll
    for (int j = 0; j < 8; ++j) rinv_a[j] = __shfl(rinv, j + 8 * hi);

    __syncthreads();                        // W1^T visible

    // ---- q = r @ W1 : outer K (two 2-VGPR A fragments live), inner 16 N-tiles
    const float* rrow = r + (size_t)(rowBase + lane16) * D_DIM + 2 * hi;
    v8f q[16];
#pragma unroll
    for (int nt = 0; nt < 16; ++nt) q[nt] = (v8f){0.f,0.f,0.f,0.f,0.f,0.f,0.f,0.f};
#pragma unroll
    for (int kk = 0; kk < 32; kk += 2) {
        v2f a0 = *(const v2f*)(rrow + 4 * kk);
        v2f a1 = *(const v2f*)(rrow + 4 * kk + 4);
#pragma unroll
        for (int nt = 0; nt < 16; ++nt) {
            const float* bb = &lds_w1t[(nt * 16 + lane16) * W1T_STRIDE + 2 * hi + 4 * kk];
            v2f b0 = *(const v2f*)(bb);         // pairs into ds_load_2addr_b64
            v2f b1 = *(const v2f*)(bb + 4);
            q[nt] = __builtin_amdgcn_wmma_f32_16x16x4_f32(
                false, a0, false, b0, (short)0, q[nt], false, false);
            q[nt] = __builtin_amdgcn_wmma_f32_16x16x4_f32(
                false, a1, false, b1, (short)0, q[nt], false, false);
        }
    }

    // ---- per-lane constants for the iteration loop
    float p1v[16], w2v[16];
#pragma unroll
    for (int nt = 0; nt < 16; ++nt) {
        p1v[nt] = p1[nt * 16 + lane16];
        w2v[nt] = w2[nt * 16 + lane16];
    }
    const float b2v  = b2[0];
    const int  niter = *n_iter_p;

    // beta = alpha * rinv in C-tile row layout (row j lanes 0-15, row j+8 lanes 16-31):
    //   x = p1 + beta*q_raw  (== p1 + alpha*(r_norm@W1));  beta += omega*rinv
    float beta[8];
#pragma unroll
    for (int j = 0; j < 8; ++j) beta[j] = 0.0f;

    for (int it = 0; it < niter; ++it) {
        float partial[8];
#pragma unroll
        for (int j = 0; j < 8; ++j) partial[j] = 0.0f;
#pragma unroll
        for (int nt = 0; nt < 16; ++nt) {
#pragma unroll
            for (int j = 0; j < 8; ++j) {
                float x = fmaf(beta[j], q[nt][j], p1v[nt]);    // (pivot + a*rn)@W1 + b1
                partial[j] = fmaf(fast_tanh(x), w2v[nt], partial[j]);
            }
        }
#pragma unroll
        for (int j = 0; j < 8; ++j) {       // reduce 16 lanes of each half -> row dot
            float sred = partial[j];
            sred += __shfl_xor(sred, 1);
            sred += __shfl_xor(sred, 2);
            sred += __shfl_xor(sred, 4);
            sred += __shfl_xor(sred, 8);
            float omega = 0.05f * (1.0f + fast_tanh(sred + b2v));
            beta[j] = fmaf(omega, rinv_a[j], beta[j]);
        }
    }

    // ---- per-row scale f = sigmoid(s) * alpha / ||r||  == sigmoid(s) * beta
    if (lane16 == 0) {                      // lanes 0 and 16 of each wave
#pragma unroll
        for (int j = 0; j < 8; ++j) {
            int lrow = wave * 16 + j + 8 * hi;
            float sv  = s[blockIdx.x * ROWS_PER_BLOCK + lrow];
            float sig = __fdividef(1.0f, 1.0f + __expf(-sv));
            lds_f[lrow] = sig * beta[j];
        }
    }
    __syncthreads();

    // ---- coalesced float4 output: out = pivot + f[row] * r  (r is cache-hot)
    const float4* r4 = (const float4*)r;
    float4*       o4 = (float4*)out;
    const int base4 = blockIdx.x * ROWS_PER_BLOCK * (D_DIM / 4);
#pragma unroll
    for (int i = 0; i < 16; ++i) {
        int idx  = tid + i * THREADS;       // 0..4095
        int lrow = idx >> 5;
        int c4   = idx & 31;
        float f  = lds_f[lrow];
        float4 rv = r4[base4 + idx];
        const float4 pv = *(const float4*)&lds_pivot[c4 * 4];
        float4 ov;
        ov.x = fmaf(f, rv.x, pv.x);
        ov.y = fmaf(f, rv.y, pv.y);
        ov.z = fmaf(f, rv.z, pv.z);
        ov.w = fmaf(f, rv.w, pv.w);
        o4[base4 + idx] = ov;
    }
}

extern "C" void kernel_launch(void* const* d_in, const int* in_sizes, int n_in,
                              void* d_out, int out_size, void* d_ws, size_t ws_size,
                              hipStream_t stream) {
    const float* r     = (const float*)d_in[0];
    const float* s     = (const float*)d_in[1];
    const float* pivot = (const float*)d_in[2];
    const float* W1    = (const float*)d_in[3];
    const float* b1    = (const float*)d_in[4];
    const float* w2    = (const float*)d_in[5];
    const float* b2    = (const float*)d_in[6];
    const int*   n_it  = (const int*)d_in[7];
    float* out = (float*)d_out;
    float* p1  = (float*)d_ws;              // 256 floats of scratch

    const int B = in_sizes[0] / D_DIM;      // 32768

    rm_p1_kernel<<<1, H_DIM, 0, stream>>>(pivot, W1, b1, p1);
    rm_main_kernel<<<B / ROWS_PER_BLOCK, THREADS, 0, stream>>>(
        r, s, pivot, W1, w2, b2, n_it, p1, out);
}